// DDCConv1D_44908178047737
// MI455X (gfx1250) — compile-verified
//
#include <hip/hip_runtime.h>
#include <math.h>

// Problem constants (fixed by the reference).
#define BB   2
#define SS   2048
#define CC   64
#define FF   64
#define KK   3
#define NN   (FF * KK)        // 192  : GEMM N dimension (f*K + k)
#define MM   (BB * SS)        // 4096 : GEMM M dimension (b*S + s)

typedef __attribute__((ext_vector_type(2))) float v2f;
typedef __attribute__((ext_vector_type(8))) float v8f;

// ---------------------------------------------------------------------------
// Phase 1: g[m, n] = sum_c x[m, c] * kernel[f, c, k],  with n = f*KK + k.
// One wave32 per 16x16 tile of g; reduction C=64 as 16x v_wmma_f32_16x16x4_f32.
// ---------------------------------------------------------------------------
__global__ __launch_bounds__(32)
void ddc_pregemm_wmma(const float* __restrict__ x,
                      const float* __restrict__ kern,
                      float* __restrict__ g) {
    const int tileN = blockIdx.x;        // 0..11
    const int tileM = blockIdx.y;        // 0..255
    const int lane  = threadIdx.x;       // 0..31 (wave32, EXEC all ones)
    const int low   = lane & 15;
    const int half  = lane >> 4;         // 0 or 1

    const int m0 = tileM * 16;
    const int n0 = tileN * 16;

    // B-matrix: this lane owns column (n0 + low) of the kernel matrix.
    const int n = n0 + low;
    const int f = n / KK;
    const int k = n - f * KK;
    const float* __restrict__ kcol = kern + (size_t)f * (CC * KK) + k;

    // A-matrix: this lane owns row (m0 + low) of x.
    const float* __restrict__ xrow = x + (size_t)(m0 + low) * CC;

    v8f acc = {};
#pragma unroll
    for (int c0 = 0; c0 < CC; c0 += 4) {
        const int cA = c0 + 2 * half;    // this lane's K-subpair within the step
        v2f a, b;
        a.x = xrow[cA + 0];              // A[row, cA],  A[row, cA+1]
        a.y = xrow[cA + 1];
        b.x = kcol[(size_t)(cA + 0) * KK];   // B[cA, col] = kernel[f, cA, k]
        b.y = kcol[(size_t)(cA + 1) * KK];
        // D = A(16x4,f32) * B(4x16,f32) + C   -> v_wmma_f32_16x16x4_f32
        acc = __builtin_amdgcn_wmma_f32_16x16x4_f32(
            /*neg_a=*/false, a, /*neg_b=*/false, b,
            /*c_mod=*/(short)0, acc, /*reuse_a=*/false, /*reuse_b=*/false);
    }

    // D layout: VGPR r, lane l -> D[r + 8*(l>>4), l&15].
    float* __restrict__ grow = g + (size_t)m0 * NN + n0;
#pragma unroll
    for (int r = 0; r < 8; ++r) {
        grow[(size_t)(r + 8 * half) * NN + low] = acc[r];
    }
}

// ---------------------------------------------------------------------------
// Phase 2: y[b,s,f] = sum_k lerp(g[b, i0c, f, k], g[b, i1c, f, k], w)
// exactly reproducing the reference's sigmoid/floor/clip arithmetic in fp32.
// ---------------------------------------------------------------------------
__global__ __launch_bounds__(256)
void ddc_gather_lerp(const float* __restrict__ g,
                     const float* __restrict__ ow,
                     const int*   __restrict__ dil_p,
                     float* __restrict__ y) {
    const int idx = blockIdx.x * 256 + threadIdx.x;   // b*S*F + s*F + f
    if (idx >= BB * SS * FF) return;
    const int f = idx % FF;
    const int s = (idx / FF) % SS;
    const int b = idx / (FF * SS);
    const int dil = dil_p[0];

    const float max_offset = 0.5f * (float)SS / (float)(dil * KK);
    const float sig = 1.0f / (1.0f + expf(-ow[f]));
    const float off = -sig * max_offset;

    const float* __restrict__ gb = g + (size_t)b * SS * NN;
    float acc = 0.0f;
#pragma unroll
    for (int k = 0; k < KK; ++k) {
        const float pos = (float)s - (float)(k * dil) + off;
        const float i0f = floorf(pos);
        const float w   = pos - i0f;
        const int   i0  = (int)i0f;
        const int i0c = min(max(i0, 0), SS - 1);
        const int i1c = min(max(i0 + 1, 0), SS - 1);
        const float g0 = gb[(size_t)i0c * NN + f * KK + k];
        const float g1 = gb[(size_t)i1c * NN + f * KK + k];
        acc += g0 + w * (g1 - g0);
    }
    y[idx] = acc;
}

// ---------------------------------------------------------------------------
// Fallback (no workspace): direct per-output computation, no WMMA. Only used
// if the harness workspace is smaller than the 3 MB intermediate.
// ---------------------------------------------------------------------------
__global__ __launch_bounds__(256)
void ddc_direct(const float* __restrict__ x,
                const float* __restrict__ kern,
                const float* __restrict__ ow,
                const int*   __restrict__ dil_p,
                float* __restrict__ y) {
    const int idx = blockIdx.x * 256 + threadIdx.x;
    if (idx >= BB * SS * FF) return;
    const int f = idx % FF;
    const int s = (idx / FF) % SS;
    const int b = idx / (FF * SS);
    const int dil = dil_p[0];

    const float max_offset = 0.5f * (float)SS / (float)(dil * KK);
    const float sig = 1.0f / (1.0f + expf(-ow[f]));
    const float off = -sig * max_offset;

    const float* __restrict__ xb = x + (size_t)b * SS * CC;
    float acc = 0.0f;
    for (int k = 0; k < KK; ++k) {
        const float pos = (float)s - (float)(k * dil) + off;
        const float i0f = floorf(pos);
        const float w   = pos - i0f;
        const int   i0  = (int)i0f;
        const int i0c = min(max(i0, 0), SS - 1);
        const int i1c = min(max(i0 + 1, 0), SS - 1);
        const float* __restrict__ x0 = xb + (size_t)i0c * CC;
        const float* __restrict__ x1 = xb + (size_t)i1c * CC;
        const float* __restrict__ kr = kern + (size_t)f * (CC * KK) + k;
        for (int c = 0; c < CC; ++c) {
            const float v = x0[c] + w * (x1[c] - x0[c]);
            acc += v * kr[(size_t)c * KK];
        }
    }
    y[idx] = acc;
}

extern "C" void kernel_launch(void* const* d_in, const int* in_sizes, int n_in,
                              void* d_out, int out_size, void* d_ws, size_t ws_size,
                              hipStream_t stream) {
    const float* x    = (const float*)d_in[0];   // (B, S, C)
    const float* kern = (const float*)d_in[1];   // (F, C, K)
    const float* ow   = (const float*)d_in[2];   // (F,)
    const int*   dil  = (const int*)d_in[3];     // scalar
    float* y = (float*)d_out;                    // (B, S, F)

    const size_t g_bytes = (size_t)MM * NN * sizeof(float);  // 3 MB

    if (ws_size >= g_bytes) {
        float* g = (float*)d_ws;
        // Phase 1: WMMA pre-GEMM, grid exactly tiles 4096x192 (no divergence).
        dim3 grid1(NN / 16, MM / 16);   // (12, 256)
        ddc_pregemm_wmma<<<grid1, 32, 0, stream>>>(x, kern, g);
        // Phase 2: gather + lerp.
        const int total = BB * SS * FF;
        ddc_gather_lerp<<<(total + 255) / 256, 256, 0, stream>>>(g, ow, dil, y);
    } else {
        const int total = BB * SS * FF;
        ddc_direct<<<(total + 255) / 256, 256, 0, stream>>>(x, kern, ow, dil, y);
    }
}